// MDTA_59287728554577
// MI455X (gfx1250) — compile-verified
//
#include <hip/hip_runtime.h>
#include <hip/hip_bf16.h>

// ---------------------------------------------------------------------------
// MDTA (Restormer transposed channel attention) for MI455X / gfx1250, fp32.
// Pipeline (fused, bandwidth-optimized):
//   1) qkv = qkv_w @ x            (1x1 conv, WMMA f32 16x16x4)
//   2) depthwise 3x3 conv + fused per-channel sum-of-squares (global f32
//      atomics) -> norms come for free, no separate 400MB read pass
//   3) attn_raw = q @ k^T         (split-K WMMA, 48x48 per wave, LDS
//      ds_add_f32 block reduction, then global atomics)
//   4) scaled softmax             (inverse norms + softplus(temp) inline)
//   5) out = attn @ v             (WMMA, v read exactly once)
//   6) y = proj_w @ out           (1x1 conv, WMMA)
// All strides compile-time -> pointer-bump + immediate-offset loads;
// streaming operands prefetched with global_prefetch_b8.
// ---------------------------------------------------------------------------

typedef float v2f __attribute__((ext_vector_type(2)));
typedef float v4f __attribute__((ext_vector_type(4)));
typedef float v8f __attribute__((ext_vector_type(8)));

static __device__ __forceinline__ v8f wmma4(v2f a, v2f b, v8f c) {
  // V_WMMA_F32_16X16X4_F32 : D = A(16x4) * B(4x16) + C(16x16), all fp32
  return __builtin_amdgcn_wmma_f32_16x16x4_f32(
      false, a, false, b, (short)0, c, false, false);
}

#define NPIX 65536   // 256*256
#define CDIM 384
#define OC3  1152
#define NB   2
#define HDIM 48

// ---------------------------------------------------------------------------
// Kernel 1/6: pointwise (1x1 conv) GEMM  Y[b,m,n] = sum_c W[m,c] * X[b,c,n]
// Block: 256 thr = 8 waves; block tile M=32 x N=256; wave tile 16 x 64.
// ---------------------------------------------------------------------------
template <int K, int N>
__global__ __launch_bounds__(256) void pw_gemm_wmma(const float* __restrict__ W,
                                                    const float* __restrict__ X,
                                                    float* __restrict__ Y,
                                                    int M) {
  const int lane = threadIdx.x & 31;
  const int wv   = threadIdx.x >> 5;
  const int m0   = blockIdx.y * 32 + (wv >> 2) * 16;
  const int n0   = blockIdx.x * 256 + (wv & 3) * 64;
  const int lr   = lane & 15;
  const int koff = (lane >> 4) << 1;        // 0 (lanes 0-15) or 2 (lanes 16-31)

  const float* wp = W + (size_t)(m0 + lr) * K + koff;
  const float* xp = X + (size_t)blockIdx.z * K * N + (size_t)koff * N + n0 + lr;

  v8f acc0 = {}, acc1 = {}, acc2 = {}, acc3 = {};

#pragma unroll 2
  for (int c0 = 0; c0 < K; c0 += 4) {
    __builtin_prefetch(xp + (size_t)16 * N, 0, 0);       // 4 K-steps ahead
    __builtin_prefetch(xp + (size_t)16 * N + 32, 0, 0);  // second 128B line
    v2f a = *(const v2f*)wp;                             // A frag (8B aligned)
    v2f b0 = { xp[0],  xp[N]      };
    v2f b1 = { xp[16], xp[N + 16] };
    v2f b2 = { xp[32], xp[N + 32] };
    v2f b3 = { xp[48], xp[N + 48] };
    acc0 = wmma4(a, b0, acc0);
    acc1 = wmma4(a, b1, acc1);
    acc2 = wmma4(a, b2, acc2);
    acc3 = wmma4(a, b3, acc3);
    wp += 4;
    xp += (size_t)4 * N;
  }

  float* yp = Y + (size_t)blockIdx.z * M * N
            + (size_t)(m0 + (lane >> 4) * 8) * N + n0 + lr;
#pragma unroll
  for (int r = 0; r < 8; ++r) {
    yp[ 0] = acc0[r];
    yp[16] = acc1[r];
    yp[32] = acc2[r];
    yp[48] = acc3[r];
    yp += N;
  }
}

// ---------------------------------------------------------------------------
// Kernel 2: depthwise 3x3 (stride 1, pad 1), 4 pixels per thread along x,
// fused per-channel sum-of-squares for q,k norm (channels < 768).
// Block = 256 threads x 4 px each
// Block covers 1024 consecutive pixels = 4 y-rows of one channel.
// ---------------------------------------------------------------------------
__global__ __launch_bounds__(256) void dwconv3x3(const float* __restrict__ X,
                                                 const float* __restrict__ Wd,
                                                 float* __restrict__ Y,
                                                 float* __restrict__ normacc) {
  const size_t pidx = ((size_t)blockIdx.x * 256 + threadIdx.x) * 4;  // first of 4 px
  const int x0 = (int)(pidx & 255);         // 0,4,...,252
  const int yh = (int)((pidx >> 8) & 255);
  const size_t cb = pidx >> 16;             // b*1152 + ch  (constant per block)
  const int ch = (int)(cb % OC3);
  const float* w = Wd + (size_t)ch * 9;
  const float* base = X + (cb << 16);

  float s[4] = {0.f, 0.f, 0.f, 0.f};
#pragma unroll
  for (int dy = -1; dy <= 1; ++dy) {
    const int yy = yh + dy;
    if ((unsigned)yy >= 256u) continue;
    const float* rowp = base + (yy << 8) + x0;
    const v4f mid = *(const v4f*)rowp;                    // x0..x0+3
    const float lf = (x0 > 0)   ? rowp[-1] : 0.f;
    const float rt = (x0 < 252) ? rowp[4]  : 0.f;
    const float w0 = w[(dy + 1) * 3 + 0];
    const float w1 = w[(dy + 1) * 3 + 1];
    const float w2 = w[(dy + 1) * 3 + 2];
    s[0] += w0 * lf     + w1 * mid.x + w2 * mid.y;
    s[1] += w0 * mid.x  + w1 * mid.y + w2 * mid.z;
    s[2] += w0 * mid.y  + w1 * mid.z + w2 * mid.w;
    s[3] += w0 * mid.z  + w1 * mid.w + w2 * rt;
  }
  v4f outv = { s[0], s[1], s[2], s[3] };
  *(v4f*)(Y + pidx) = outv;

  // fused sum-of-squares for q (ch<384) and k (384<=ch<768) channels
  __shared__ float red[256];
  red[threadIdx.x] = s[0]*s[0] + s[1]*s[1] + s[2]*s[2] + s[3]*s[3];
  __syncthreads();
  for (int off = 128; off > 0; off >>= 1) {
    if (threadIdx.x < off) red[threadIdx.x] += red[threadIdx.x + off];
    __syncthreads();
  }
  if (threadIdx.x == 0 && ch < 768) {
    const int b = (int)(cb / OC3);
    atomicAdd(normacc + (size_t)b * 768 + ch, red[0]);
  }
}

// ---------------------------------------------------------------------------
// Kernel 3: attn_raw[bh] (48x48) = q @ k^T, split-K over n.
// Each wave computes the FULL 48x48 (9 accumulators) for a 512-wide n chunk
// -> q,k each read exactly once.  Partials combined in LDS with ds_add_f32,
// then one global atomicAdd per element per block.
// grid (16 bh, 16 chunk groups), 256 thr = 8 waves.
// ---------------------------------------------------------------------------
__global__ __launch_bounds__(256) void attn_qk_wmma(const float* __restrict__ QKV,
                                                    float* __restrict__ attn) {
  const int lane = threadIdx.x & 31;
  const int wv   = threadIdx.x >> 5;
  const int bh = blockIdx.x;
  const int b = bh >> 3, h = bh & 7;
  const int lr = lane & 15, koff = (lane >> 4) << 1;

  __shared__ float lred[HDIM * HDIM];       // 2304 floats = 9KB
  for (int i = threadIdx.x; i < HDIM * HDIM; i += 256) lred[i] = 0.f;
  __syncthreads();

  const int nstart = (blockIdx.y * 8 + wv) * 512;

  const float* qp  = QKV + ((size_t)b * OC3 +       h * HDIM + lr) * (size_t)NPIX
                   + nstart + koff;
  const float* kp  = QKV + ((size_t)b * OC3 + 384 + h * HDIM + lr) * (size_t)NPIX
                   + nstart + koff;
  const float* qp2 = qp + (size_t)32 * NPIX;   // 8MB > imm24 -> own pointer
  const float* kp2 = kp + (size_t)32 * NPIX;

  v8f acc[3][3] = {};
  for (int o = 0; o < 16; ++o) {            // 16 x (8 x 4) = 512 n
    __builtin_prefetch(qp  + 64, 0, 0);     // 128B group two iterations ahead per stream
    __builtin_prefetch(qp  + (size_t)16 * NPIX + 64, 0, 0);
    __builtin_prefetch(qp2 + 64, 0, 0);
    __builtin_prefetch(kp  + 64, 0, 0);
    __builtin_prefetch(kp  + (size_t)16 * NPIX + 64, 0, 0);
    __builtin_prefetch(kp2 + 64, 0, 0);
#pragma unroll
    for (int u = 0; u < 8; ++u) {
      v2f a0 = *(const v2f*)qp;
      v2f a1 = *(const v2f*)(qp + (size_t)16 * NPIX);   // 4MB imm ok
      v2f a2 = *(const v2f*)qp2;
      v2f b0 = *(const v2f*)kp;
      v2f b1 = *(const v2f*)(kp + (size_t)16 * NPIX);
      v2f b2 = *(const v2f*)kp2;
      acc[0][0] = wmma4(a0, b0, acc[0][0]);
      acc[0][1] = wmma4(a0, b1, acc[0][1]);
      acc[0][2] = wmma4(a0, b2, acc[0][2]);
      acc[1][0] = wmma4(a1, b0, acc[1][0]);
      acc[1][1] = wmma4(a1, b1, acc[1][1]);
      acc[1][2] = wmma4(a1, b2, acc[1][2]);
      acc[2][0] = wmma4(a2, b0, acc[2][0]);
      acc[2][1] = wmma4(a2, b1, acc[2][1]);
      acc[2][2] = wmma4(a2, b2, acc[2][2]);
      qp += 4; qp2 += 4; kp += 4; kp2 += 4;
    }
  }

  // combine the block's 8 waves in LDS (ds_add_f32)
#pragma unroll
  for (int ti = 0; ti < 3; ++ti) {
#pragma unroll
    for (int tj = 0; tj < 3; ++tj) {
      const int base = (ti * 16 + (lane >> 4) * 8) * HDIM + tj * 16 + lr;
#pragma unroll
      for (int r = 0; r < 8; ++r)
        atomicAdd(&lred[base + r * HDIM], acc[ti][tj][r]);
    }
  }
  __syncthreads();

  float* gp = attn + (size_t)bh * HDIM * HDIM;
  for (int i = threadIdx.x; i < HDIM * HDIM; i += 256)
    atomicAdd(gp + i, lred[i]);
}

// ---------------------------------------------------------------------------
// Kernel 4: logits = raw * invq[c] * invk[d] * (softplus(log_temp)+1e-6),
// row softmax.  Inverse norms computed inline from normacc.
// One block per (b,h), threads 0..47 own one row each.
// ---------------------------------------------------------------------------
__global__ void softmax_attn(float* __restrict__ attn,
                             const float* __restrict__ normacc,
                             const float* __restrict__ log_temp) {
  const int bh = blockIdx.x;
  const int b = bh >> 3, h = bh & 7;
  const int r = threadIdx.x;

  __shared__ float invq_sh[HDIM], invk_sh[HDIM];
  if (r < HDIM) {
    invq_sh[r] = 1.f / fmaxf(sqrtf(normacc[(size_t)b * 768 +       h * HDIM + r]), 1e-12f);
    invk_sh[r] = 1.f / fmaxf(sqrtf(normacc[(size_t)b * 768 + 384 + h * HDIM + r]), 1e-12f);
  }
  __syncthreads();
  if (r >= HDIM) return;

  const float lt = log_temp[h];
  const float temp = (lt > 20.f ? lt : log1pf(expf(lt))) + 1e-6f;  // softplus
  float* row = attn + ((size_t)bh * HDIM + r) * HDIM;
  const float sq = invq_sh[r] * temp;

  float vals[HDIM];
  float mx = -3.4e38f;
  for (int d = 0; d < HDIM; ++d) {
    const float v = row[d] * sq * invk_sh[d];
    vals[d] = v;
    mx = fmaxf(mx, v);
  }
  float sum = 0.f;
  for (int d = 0; d < HDIM; ++d) {
    const float e = expf(vals[d] - mx);
    vals[d] = e;
    sum += e;
  }
  const float rs = 1.f / sum;
  for (int d = 0; d < HDIM; ++d) row[d] = vals[d] * rs;
}

// ---------------------------------------------------------------------------
// Kernel 5: out[bh, c, n] = sum_d attn[c,d] * v[d,n].  One wave computes all
// 3 c-tiles (48 rows) of one 16-wide n column -> v read exactly once.
// grid (512 n-groups, 16 bh).
// ---------------------------------------------------------------------------
__global__ __launch_bounds__(256) void attn_v_wmma(const float* __restrict__ attn,
                                                   const float* __restrict__ QKV,
                                                   float* __restrict__ Out) {
  const int lane = threadIdx.x & 31;
  const int wv   = threadIdx.x >> 5;
  const int n0 = (blockIdx.x * 8 + wv) * 16;
  const int bh = blockIdx.y;
  const int b = bh >> 3, h = bh & 7;
  const int lr = lane & 15, koff = (lane >> 4) << 1;

  const float* ap = attn + (size_t)bh * HDIM * HDIM + (size_t)lr * HDIM + koff;
  const float* vp = QKV + ((size_t)b * OC3 + 768 + h * HDIM + koff) * (size_t)NPIX
                  + n0 + lr;

  v8f acc0 = {}, acc1 = {}, acc2 = {};
#pragma unroll
  for (int k0 = 0; k0 < HDIM; k0 += 4) {
    v2f bf = { vp[0], vp[NPIX] };
    v2f a0 = *(const v2f*)ap;                 // rows 0..15
    v2f a1 = *(const v2f*)(ap + 16 * HDIM);   // rows 16..31 (3072B imm)
    v2f a2 = *(const v2f*)(ap + 32 * HDIM);   // rows 32..47 (6144B imm)
    acc0 = wmma4(a0, bf, acc0);
    acc1 = wmma4(a1, bf, acc1);
    acc2 = wmma4(a2, bf, acc2);
    ap += 4;
    vp += (size_t)4 * NPIX;
  }

  float* op = Out + ((size_t)b * CDIM + h * HDIM + (lane >> 4) * 8) * (size_t)NPIX
            + n0 + lr;
#pragma unroll
  for (int r = 0; r < 8; ++r) { op[0] = acc0[r]; op += NPIX; }
  op += (size_t)8 * NPIX;                     // skip to rows 16+hi*8
#pragma unroll
  for (int r = 0; r < 8; ++r) { op[0] = acc1[r]; op += NPIX; }
  op += (size_t)8 * NPIX;
#pragma unroll
  for (int r = 0; r < 8; ++r) { op[0] = acc2[r]; op += NPIX; }
}

// ---------------------------------------------------------------------------
extern "C" void kernel_launch(void* const* d_in, const int* in_sizes, int n_in,
                              void* d_out, int out_size, void* d_ws, size_t ws_size,
                              hipStream_t stream) {
  (void)in_sizes; (void)n_in; (void)out_size; (void)ws_size;
  const float* x        = (const float*)d_in[0];  // (2,384,256,256)
  const float* qkv_w    = (const float*)d_in[1];  // (1152,384)
  const float* dw_w     = (const float*)d_in[2];  // (1152,1,3,3)
  const float* proj_w   = (const float*)d_in[3];  // (384,384)
  const float* log_temp = (const float*)d_in[4];  // (8,1,1)

  const size_t q1 = (size_t)NB * OC3 * NPIX * sizeof(float);  // 604 MB

  char* ws = (char*)d_ws;
  float* qkv     = (float*)ws;                       // [B,1152,N]
  float* dwq     = (float*)(ws + q1);                // [B,1152,N]
  float* normacc = (float*)(ws + 2 * q1);            // [B,768] sum of squares
  float* attn    = (float*)(ws + 2 * q1 + 6144);     // [16,48,48]
  float* aout    = qkv;                              // reuse qkv region

  // 1) qkv = qkv_w @ x
  pw_gemm_wmma<CDIM, NPIX><<<dim3(NPIX / 256, OC3 / 32, NB), 256, 0, stream>>>(
      qkv_w, x, qkv, OC3);

  // zero norm accumulators + attn partials (contiguous)
  hipMemsetAsync(normacc, 0,
                 (size_t)NB * 768 * sizeof(float) +
                 (size_t)16 * HDIM * HDIM * sizeof(float) + 0, stream);

  // 2) depthwise 3x3 + fused q,k sum-of-squares
  dwconv3x3<<<(unsigned)(((size_t)NB * OC3 * NPIX) / 1024), 256, 0, stream>>>(
      qkv, dw_w, dwq, normacc);

  // 3) attn_raw = q @ k^T (split-K, LDS ds_add_f32 reduce + global atomics)
  attn_qk_wmma<<<dim3(16, 16), 256, 0, stream>>>(dwq, attn);

  // 4) scaled softmax (inverse norms inline)
  softmax_attn<<<16, 64, 0, stream>>>(attn, normacc, log_temp);

  // 5) out = attn @ v
  attn_v_wmma<<<dim3(NPIX / 128, 16), 256, 0, stream>>>(attn, dwq, aout);

  // 6) y = proj_w @ out
  pw_gemm_wmma<CDIM, NPIX><<<dim3(NPIX / 256, CDIM / 32, NB), 256, 0, stream>>>(
      proj_w, aout, (float*)d_out, CDIM);
}